// Attention_1_81612968559058
// MI455X (gfx1250) — compile-verified
//
#include <hip/hip_runtime.h>

typedef float v2f __attribute__((ext_vector_type(2)));
typedef float v8f __attribute__((ext_vector_type(8)));

#define NB 64
#define LD 512
#define TD 128
#define DD 1024

// ---------------------------------------------------------------------------
// Kernel 1: s1[n,l] = dot(f1[n,l,:], w)   (32768 rows of 1024)
// One wave per row; float4 coalesced loads; wave32 shuffle reduction.
// ---------------------------------------------------------------------------
__launch_bounds__(256)
__global__ void k_rowdot(const float* __restrict__ f1,
                         const float* __restrict__ w,
                         float* __restrict__ s1) {
    int row  = blockIdx.x * 8 + (threadIdx.x >> 5);
    int lane = threadIdx.x & 31;
    const float4* r4 = (const float4*)(f1 + (size_t)row * DD);
    const float4* w4 = (const float4*)w;
    float acc = 0.0f;
#pragma unroll
    for (int i = 0; i < 8; ++i) {
        float4 a = r4[lane + 32 * i];
        float4 b = w4[lane + 32 * i];
        acc += a.x * b.x + a.y * b.y + a.z * b.z + a.w * b.w;
    }
#pragma unroll
    for (int off = 16; off; off >>= 1) acc += __shfl_xor(acc, off, 32);
    if (lane == 0) s1[row] = acc;
}

// ---------------------------------------------------------------------------
// Kernel 2: p[n,:] = softmax(s1[n,:]) over L=512. One 256-thread block per n.
// (s2[n,t]+b is constant along l and cancels in softmax -> exact.)
// ---------------------------------------------------------------------------
__launch_bounds__(256)
__global__ void k_softmax(const float* __restrict__ s1,
                          float* __restrict__ p) {
    __shared__ float red[9];
    int n   = blockIdx.x;
    int tid = threadIdx.x;
    float v0 = s1[n * LD + tid];
    float v1 = s1[n * LD + tid + 256];

    float m = fmaxf(v0, v1);
#pragma unroll
    for (int off = 16; off; off >>= 1) m = fmaxf(m, __shfl_xor(m, off, 32));
    if ((tid & 31) == 0) red[tid >> 5] = m;
    __syncthreads();
    if (tid == 0) {
        float t = red[0];
#pragma unroll
        for (int i = 1; i < 8; ++i) t = fmaxf(t, red[i]);
        red[8] = t;
    }
    __syncthreads();
    m = red[8];

    float e0 = expf(v0 - m);
    float e1 = expf(v1 - m);
    float s = e0 + e1;
#pragma unroll
    for (int off = 16; off; off >>= 1) s += __shfl_xor(s, off, 32);
    __syncthreads();
    if ((tid & 31) == 0) red[tid >> 5] = s;
    __syncthreads();
    if (tid == 0) {
        float t = red[0];
#pragma unroll
        for (int i = 1; i < 8; ++i) t += red[i];
        red[8] = t;
    }
    __syncthreads();
    float inv = 1.0f / red[8];

    p[n * LD + tid]       = e0 * inv;
    p[n * LD + tid + 256] = e1 * inv;
}

// ---------------------------------------------------------------------------
// Kernel 3: att[n,t,l] = p[n,l]  (broadcast over T), float4 stores.
// ---------------------------------------------------------------------------
__launch_bounds__(256)
__global__ void k_att_bcast(const float* __restrict__ p,
                            float* __restrict__ att) {
    int idx = blockIdx.x * 256 + threadIdx.x;   // over N*T*L/4 = 1048576
    int l4  = idx & 127;                        // L/4 = 128
    int n   = idx >> 14;                        // / (T * L/4)
    float4 v = ((const float4*)p)[n * 128 + l4];
    ((float4*)att)[idx] = v;
}

// ---------------------------------------------------------------------------
// Kernel 4: f_hat[n,t,d] = sum_l p[n,l] * f1[n,l,d]   via V_WMMA_F32_16X16X4_F32
// Block = 256 threads = 8 waves; block covers (n, 128-wide d slice); each wave
// owns a 16-wide d slice. f1 rows staged into LDS in 32-row chunks with b128
// loads. A-matrix rows are all == p, so the 16x16 C tile holds r[d:d+16]
// duplicated; each register half-pair gives 2 output rows per b32 store.
// ---------------------------------------------------------------------------
__launch_bounds__(256)
__global__ void k_wmma_fhat(const float* __restrict__ f1,
                            const float* __restrict__ p,
                            float* __restrict__ fhat) {
    __shared__ float Bs[32][132];   // 32 rows x 128 cols, +4 pad
    __shared__ float Ps[LD];

    int blk   = blockIdx.x;         // n*8 + dblk
    int n     = blk >> 3;
    int d0    = (blk & 7) * 128;
    int tid   = threadIdx.x;
    int wave  = tid >> 5;
    int lane  = tid & 31;
    int laneN = lane & 15;
    int hi    = lane >> 4;          // 0: lanes 0-15, 1: lanes 16-31

    Ps[tid]       = p[n * LD + tid];
    Ps[tid + 256] = p[n * LD + tid + 256];

    const float* f1n = f1 + (size_t)n * LD * DD;
    int dw = wave * 16;

    v8f c = {};
    for (int stage = 0; stage < 16; ++stage) {
        __syncthreads();
        int l0 = stage * 32;
        // stage 32 rows x 128 floats (16 KB) cooperatively, b128 loads
#pragma unroll
        for (int i = 0; i < 4; ++i) {
            int e  = tid + 256 * i;          // 0..1023 float4 slots
            int r  = e >> 5;                 // row 0..31
            int c4 = e & 31;                 // float4 column
            float4 v = *(const float4*)(f1n + (size_t)(l0 + r) * DD + d0 + c4 * 4);
            *(float4*)&Bs[r][c4 * 4] = v;
        }
        __syncthreads();
#pragma unroll
        for (int k = 0; k < 8; ++k) {
            int lr = k * 4 + hi * 2;         // K=0/2 in v0, K=1/3 in v1
            v2f a, b;
            a.x = Ps[l0 + lr];
            a.y = Ps[l0 + lr + 1];
            b.x = Bs[lr][dw + laneN];
            b.y = Bs[lr + 1][dw + laneN];
            c = __builtin_amdgcn_wmma_f32_16x16x4_f32(
                    false, a, false, b, (short)0, c, false, false);
        }
    }

    // c[0]: lanes 0-15 hold r[d0+dw+lane] (M=0 row), lanes 16-31 hold the
    // same values (M=8 row). Each b32 store covers 2 of the 128 t-rows.
    float val = c[0];
    float* outp = fhat + (size_t)n * TD * DD + d0 + dw + laneN;
    for (int t = hi; t < TD; t += 2) {
        outp[(size_t)t * DD] = val;
    }
}

// ---------------------------------------------------------------------------
extern "C" void kernel_launch(void* const* d_in, const int* in_sizes, int n_in,
                              void* d_out, int out_size, void* d_ws, size_t ws_size,
                              hipStream_t stream) {
    const float* f1 = (const float*)d_in[0];   // [64,512,1024]
    // d_in[1] = feature_2 : unused (cancels in softmax)
    const float* w  = (const float*)d_in[2];   // [1024]
    // d_in[3] = b : unused (cancels in softmax)

    float* fhat = (float*)d_out;                        // [64,128,1024]
    float* att  = (float*)d_out + (size_t)NB * TD * DD; // [64,128,512]

    float* s1 = (float*)d_ws;                  // 64*512 floats
    float* p  = s1 + NB * LD;                  // 64*512 floats

    k_rowdot<<<(NB * LD) / 8, 256, 0, stream>>>(f1, w, s1);
    k_softmax<<<NB, 256, 0, stream>>>(s1, p);
    k_att_bcast<<<(NB * TD * LD / 4) / 256, 256, 0, stream>>>(p, att);
    k_wmma_fhat<<<NB * 8, 256, 0, stream>>>(f1, p, fhat);
}